// TotalLoss_46557445489044
// MI455X (gfx1250) — compile-verified
//
#include <hip/hip_runtime.h>
#include <math.h>

typedef __attribute__((ext_vector_type(2))) float v2f;
typedef __attribute__((ext_vector_type(8))) float v8f;

#define N_PT 4096
#define C_F  256
#define HW   262144   // 512*512

// ---------------- deterministic block reduction (256 threads) ----------------
__device__ __forceinline__ float blk_reduce_256(float v, float* sm) {
  int tid = threadIdx.x;
  sm[tid] = v;
  __syncthreads();
  #pragma unroll
  for (int s = 128; s > 0; s >>= 1) {
    if (tid < s) sm[tid] += sm[tid + s];
    __syncthreads();
  }
  float r = sm[0];
  __syncthreads();
  return r;
}

// ---------------- score loss: masked MSE over (16,512,512) ----------------
__global__ void score_kernel(const float* __restrict__ score,
                             const float* __restrict__ label,
                             float* __restrict__ pnum, float* __restrict__ pcnt) {
  __shared__ float sm[256];
  const int total = 16 * HW;
  int stride = gridDim.x * 256;
  float num = 0.f, cnt = 0.f;
  for (int idx = blockIdx.x * 256 + threadIdx.x; idx < total; idx += stride) {
    int bs = idx / HW;
    int hw = idx - bs * HW;
    float s = __builtin_nontemporal_load(score + idx);
    float l = (bs < 8) ? __builtin_nontemporal_load(label + (bs * HW + hw) * 2)
                       : __builtin_nontemporal_load(label + ((bs - 8) * HW + hw) * 2 + 1);
    if (s > 1e-8f) { float d = s - l; num += d * d; cnt += 1.0f; }
  }
  float r;
  r = blk_reduce_256(num, sm); if (threadIdx.x == 0) pnum[blockIdx.x] = r;
  r = blk_reduce_256(cnt, sm); if (threadIdx.x == 0) pcnt[blockIdx.x] = r;
}

// ------ sinkhorn match + pose losses via V_WMMA_F32_16X16X4_F32 ------
// A (16x4): rows = output coord i (pose rows, zero-padded), K = j.
//   lane l (l16=l&15, hlf=l>>4) holds A[m=l16][k=2*hlf+v] in vgpr v.
// B (4x16): cols = 16 points, K = j.
//   lane l holds B[k=2*hlf+v][n=l16] = src[point l16][2*hlf+v].
// D (16x16): lane n (n<16) holds D[m=vgpr][n] = transformed coord m of point n.
__global__ void match_pose_kernel(const float* __restrict__ kp,
                                  const float* __restrict__ pose,
                                  const float* __restrict__ tr1,
                                  const float* __restrict__ proj1,
                                  const float* __restrict__ tr2,
                                  const float* __restrict__ proj2,
                                  float* __restrict__ pm1, float* __restrict__ pm2,
                                  float* __restrict__ pp1, float* __restrict__ pp2) {
  __shared__ float sm[256];
  int tid  = threadIdx.x;
  int wave = tid >> 5;
  int lane = tid & 31;
  int hlf  = lane >> 4;
  int l16  = lane & 15;
  int b = blockIdx.y;                         // 0..7
  int p = blockIdx.x * 128 + wave * 16 + l16; // point index 0..4095

  v2f aP = (v2f)0.f, aT1 = (v2f)0.f, aT2 = (v2f)0.f;
  if (l16 < 4) {
    int o = b * 16 + l16 * 4 + hlf * 2;
    aP.x  = pose[o]; aP.y  = pose[o + 1];
    aT1.x = tr1[o];  aT1.y = tr1[o + 1];
    aT2.x = tr2[o];  aT2.y = tr2[o + 1];
  }
  int ko = (b * N_PT + p) * 4 + hlf * 2;
  v2f braw; braw.x = kp[ko]; braw.y = kp[ko + 1];
  v2f bhom = braw;
  if (hlf) bhom.y = 1.0f;                     // homogenize w=1 (sinkhorn variant)

  v8f z = (v8f)0.f;
  v8f gth = __builtin_amdgcn_wmma_f32_16x16x4_f32(false, aP,  false, bhom, (short)0, z, false, false);
  v8f gtr = __builtin_amdgcn_wmma_f32_16x16x4_f32(false, aP,  false, braw, (short)0, z, false, false);
  v8f pr1 = __builtin_amdgcn_wmma_f32_16x16x4_f32(false, aT1, false, braw, (short)0, z, false, false);
  v8f pr2 = __builtin_amdgcn_wmma_f32_16x16x4_f32(false, aT2, false, braw, (short)0, z, false, false);

  float m1 = 0.f, m2 = 0.f, q1 = 0.f, q2 = 0.f;
  if (hlf == 0) {  // lanes 0..15 hold full result for their point
    int po = (b * N_PT + p) * 3;
    float dx = gth[0] - proj1[po], dy = gth[1] - proj1[po + 1], dz = gth[2] - proj1[po + 2];
    m1 = sqrtf(dx * dx + dy * dy + dz * dz);
    dx = gth[0] - proj2[po]; dy = gth[1] - proj2[po + 1]; dz = gth[2] - proj2[po + 2];
    m2 = sqrtf(dx * dx + dy * dy + dz * dz);
    q1 = fabsf(pr1[0] - gtr[0]) + fabsf(pr1[1] - gtr[1]) + fabsf(pr1[2] - gtr[2]);
    q2 = fabsf(pr2[0] - gtr[0]) + fabsf(pr2[1] - gtr[1]) + fabsf(pr2[2] - gtr[2]);
  }
  int pi = blockIdx.y * gridDim.x + blockIdx.x;
  float r;
  r = blk_reduce_256(m1, sm); if (tid == 0) pm1[pi] = r;
  r = blk_reduce_256(m2, sm); if (tid == 0) pm2[pi] = r;
  r = blk_reduce_256(q1, sm); if (tid == 0) pp1[pi] = r;
  r = blk_reduce_256(q2, sm); if (tid == 0) pp2[pi] = r;
}

// ---------------- gen-points loss (also WMMA transform) ----------------
__global__ void genpoints_kernel(const float* __restrict__ kp,
                                 const float* __restrict__ pose,
                                 const float* __restrict__ kgen,
                                 float* __restrict__ pgp) {
  __shared__ float sm[256];
  int tid  = threadIdx.x;
  int wave = tid >> 5, lane = tid & 31, hlf = lane >> 4, l16 = lane & 15;
  int bb = blockIdx.y;     // 0..15
  int pb = bb & 7;         // pose index (poses are concatenated)
  int p  = blockIdx.x * 128 + wave * 16 + l16;

  v2f aP = (v2f)0.f;
  if (l16 < 4) { int o = pb * 16 + l16 * 4 + hlf * 2; aP.x = pose[o]; aP.y = pose[o + 1]; }

  v2f bsrc;
  if (bb < 8) {  // src = raw key_points[bb]
    int ko = (bb * N_PT + p) * 4 + hlf * 2;
    bsrc.x = kp[ko]; bsrc.y = kp[ko + 1];
  } else {       // src = (gx, gy, 0, 1) from key_points_gen[bb-8]
    if (hlf == 0) { int go = ((bb - 8) * N_PT + p) * 2; bsrc.x = kgen[go]; bsrc.y = kgen[go + 1]; }
    else          { bsrc.x = 0.f; bsrc.y = 1.f; }
  }
  v8f z = (v8f)0.f;
  v8f s1 = __builtin_amdgcn_wmma_f32_16x16x4_f32(false, aP, false, bsrc, (short)0, z, false, false);

  float t = 0.f;
  if (hlf == 0) {
    float tx, ty;
    if (bb < 8) { int go = ((8 + bb) * N_PT + p) * 2; tx = kgen[go]; ty = kgen[go + 1]; }
    else        { int ko = (bb * N_PT + p) * 4;        tx = kp[ko];   ty = kp[ko + 1]; }
    t = fabsf(s1[0] - tx) + fabsf(s1[1] - ty);
  }
  float r = blk_reduce_256(t, sm);
  if (tid == 0) pgp[blockIdx.y * gridDim.x + blockIdx.x] = r;
}

// ---------------- relation counts ----------------
__global__ void cnt_kernel(const float* __restrict__ rel, int* __restrict__ cntb) {
  __shared__ float sm[256];
  int b = blockIdx.x, tid = threadIdx.x;
  float c = 0.f;
  for (int n = tid; n < N_PT; n += 256) {
    int o = (b * N_PT + n) * 4;                  // relation[b,n,1,{0,1}] = o+2, o+3
    if (rel[o + 2] > 0.f && rel[o + 3] > 0.f) c += 1.f;
  }
  float r = blk_reduce_256(c, sm);
  if (tid == 0) cntb[b] = (int)(r + 0.5f);
}

// ------- feature cosine losses: 402 MB streamed once, NT loads -------
__global__ void feature_kernel(const float* __restrict__ ptA, const float* __restrict__ ptB,
                               const float* __restrict__ plA, const float* __restrict__ plB,
                               const float* __restrict__ kqA, const float* __restrict__ kqB,
                               const int* __restrict__ cntb,
                               float* __restrict__ ps0, float* __restrict__ ps1,
                               float* __restrict__ ps2) {
  __shared__ float sm[256];
  int tid = threadIdx.x;
  int b = blockIdx.y;
  int n = blockIdx.x * 256 + tid;
  int base = b * C_F * N_PT + n;   // (B,C,N): lane owns column n, strides over C
  float d0 = 0, na0 = 0, nb0 = 0, d1 = 0, na1 = 0, nb1 = 0, d2 = 0, na2 = 0, nb2 = 0;
  #pragma unroll 4
  for (int c = 0; c < C_F; c++) {
    int o = base + c * N_PT;
    float a = __builtin_nontemporal_load(ptA + o), x = __builtin_nontemporal_load(ptB + o);
    d0 += a * x; na0 += a * a; nb0 += x * x;
    a = __builtin_nontemporal_load(plA + o); x = __builtin_nontemporal_load(plB + o);
    d1 += a * x; na1 += a * a; nb1 += x * x;
    a = __builtin_nontemporal_load(kqA + o); x = __builtin_nontemporal_load(kqB + o);
    d2 += a * x; na2 += a * a; nb2 += x * x;
  }
  float c0 = 1.f - d0 / fmaxf(sqrtf(na0) * sqrtf(nb0), 1e-8f);
  float c1 = 1.f - d1 / fmaxf(sqrtf(na1) * sqrtf(nb1), 1e-8f);
  float c2 = 1.f - d2 / fmaxf(sqrtf(na2) * sqrtf(nb2), 1e-8f);
  bool m = n < cntb[b];
  int pi = b * gridDim.x + blockIdx.x;
  float r;
  r = blk_reduce_256(m ? c0 : 0.f, sm); if (tid == 0) ps0[pi] = r;
  r = blk_reduce_256(m ? c1 : 0.f, sm); if (tid == 0) ps1[pi] = r;
  r = blk_reduce_256(c2, sm);           if (tid == 0) ps2[pi] = r;
}

// ---------------- finalize: deterministic serial combine ----------------
__global__ void finalize_kernel(const float* __restrict__ pose,
                                const float* __restrict__ tr1, const float* __restrict__ tr2,
                                const float* __restrict__ pnum, const float* __restrict__ pcnt,
                                const float* __restrict__ pm1, const float* __restrict__ pm2,
                                const float* __restrict__ pp1, const float* __restrict__ pp2,
                                const float* __restrict__ pgp,
                                const float* __restrict__ ps0, const float* __restrict__ ps1,
                                const float* __restrict__ ps2,
                                const int* __restrict__ cntb, float* __restrict__ out) {
  if (threadIdx.x != 0 || blockIdx.x != 0) return;
  float num = 0.f, cf = 0.f;
  for (int i = 0; i < 512; i++) { num += pnum[i]; cf += pcnt[i]; }
  float l_score = num / fmaxf(cf, 1.f);

  float m1 = 0, m2 = 0, q1 = 0, q2 = 0;
  for (int i = 0; i < 256; i++) { m1 += pm1[i]; m2 += pm2[i]; q1 += pp1[i]; q2 += pp2[i]; }
  float l_match1 = m1 / 32768.f, l_match2 = m2 / 32768.f;
  float l_pose1 = q1 / 98304.f,  l_pose2 = q2 / 98304.f;

  float l_tra1 = 0, l_rot1 = 0, l_tra2 = 0, l_rot2 = 0;
  for (int b = 0; b < 8; b++) {
    int o = b * 16;
    float dx = tr1[o + 3] - pose[o + 3], dy = tr1[o + 7] - pose[o + 7], dz = tr1[o + 11] - pose[o + 11];
    l_tra1 += sqrtf(dx * dx + dy * dy + dz * dz);
    dx = tr2[o + 3] - pose[o + 3]; dy = tr2[o + 7] - pose[o + 7]; dz = tr2[o + 11] - pose[o + 11];
    l_tra2 += sqrtf(dx * dx + dy * dy + dz * dz);
    float a2v = acosf(fminf(fmaxf(pose[o], -1.f), 1.f));
    l_rot1 += fabsf(acosf(fminf(fmaxf(tr1[o], -1.f), 1.f)) - a2v);
    l_rot2 += fabsf(acosf(fminf(fmaxf(tr2[o], -1.f), 1.f)) - a2v);
  }
  l_tra1 *= 0.125f; l_tra2 *= 0.125f;
  l_rot1 = l_rot1 * 0.125f / 3.1415f * 180.f;
  l_rot2 = l_rot2 * 0.125f / 3.1415f * 180.f;

  float cdiv = (fminf(l_rot1, l_rot2) > 0.f) ? 2.f : 1.f;
  float l_match = (l_match1 + l_match2) / cdiv;
  float l_pose  = (l_pose1 + l_pose2) / cdiv;
  float l_tra   = (l_tra1 + l_tra2) / cdiv;
  float l_rot   = (l_rot1 + l_rot2) / cdiv;

  float l_gb = 0, l_gi = 0, l_gpa = 0;
  for (int b = 0; b < 16; b++) {
    float S0 = 0, S1 = 0;
    for (int ch = 0; ch < 16; ch++) { S0 += ps0[b * 16 + ch]; S1 += ps1[b * 16 + ch]; }
    float den = fmaxf((float)cntb[b], 1.f);
    l_gb += S0 / den; l_gi += S1 / den;
  }
  l_gb *= (1.f / 16.f); l_gi *= (1.f / 16.f);
  for (int i = 0; i < 256; i++) l_gpa += ps2[i];
  l_gpa *= (1.f / 65536.f);

  float gpo = 0; for (int i = 0; i < 512; i++) gpo += pgp[i];
  float l_gpo = gpo * (1.f / 131072.f);

  out[0] = l_pose + l_score + l_match + l_tra + l_rot + l_gb + l_gi + l_gpa + l_gpo; // + l_kpl(=0)
}

extern "C" void kernel_launch(void* const* d_in, const int* in_sizes, int n_in,
                              void* d_out, int out_size, void* d_ws, size_t ws_size,
                              hipStream_t stream) {
  const float* score = (const float*)d_in[0];
  const float* label = (const float*)d_in[1];
  const float* kp    = (const float*)d_in[2];
  const float* pose  = (const float*)d_in[3];
  /* d_in[4] = batch_size (int scalar), fixed at 16 */
  const float* tr1   = (const float*)d_in[5];
  const float* proj1 = (const float*)d_in[6];
  const float* tr2   = (const float*)d_in[7];
  const float* proj2 = (const float*)d_in[8];
  const float* ptA   = (const float*)d_in[9];
  const float* ptB   = (const float*)d_in[10];
  const float* plA   = (const float*)d_in[11];
  const float* plB   = (const float*)d_in[12];
  const float* kqA   = (const float*)d_in[13];
  const float* kqB   = (const float*)d_in[14];
  const float* rel   = (const float*)d_in[15];
  const float* kgen  = (const float*)d_in[16];

  float* w   = (float*)d_ws;
  float* pnum = w;          float* pcnt = w + 512;
  float* pm1  = w + 1024;   float* pm2  = w + 1280;
  float* pp1  = w + 1536;   float* pp2  = w + 1792;
  float* pgp  = w + 2048;
  float* ps0  = w + 2560;   float* ps1  = w + 2816;   float* ps2 = w + 3072;
  int*   cntb = (int*)(w + 3328);

  score_kernel<<<512, 256, 0, stream>>>(score, label, pnum, pcnt);
  match_pose_kernel<<<dim3(32, 8), 256, 0, stream>>>(kp, pose, tr1, proj1, tr2, proj2,
                                                     pm1, pm2, pp1, pp2);
  genpoints_kernel<<<dim3(32, 16), 256, 0, stream>>>(kp, pose, kgen, pgp);
  cnt_kernel<<<16, 256, 0, stream>>>(rel, cntb);
  feature_kernel<<<dim3(16, 16), 256, 0, stream>>>(ptA, ptB, plA, plB, kqA, kqB, cntb,
                                                   ps0, ps1, ps2);
  finalize_kernel<<<1, 1, 0, stream>>>(pose, tr1, tr2, pnum, pcnt, pm1, pm2, pp1, pp2,
                                       pgp, ps0, ps1, ps2, cntb, (float*)d_out);
}